// MCPTPromptModel_63599875719684
// MI455X (gfx1250) — compile-verified
//
#include <hip/hip_runtime.h>
#include <hip/hip_bf16.h>

#define B_  32
#define S_  1024
#define D_  1024
#define P_  16
#define PID_ 50256
#define EPS_ 1e-5f

typedef __attribute__((ext_vector_type(2))) float v2f;
typedef __attribute__((ext_vector_type(8))) float v8f;

// ---------------------------------------------------------------------------
// GEMM for the prompt MLP: out[16,D] = act(A[16,D] @ W[D,D] + bias)
// One wave (32 threads) per 16-column tile of N. Uses V_WMMA_F32_16X16X4_F32.
//
// VGPR layouts (CDNA5 ISA 7.12.2, fp32):
//   A 16x4 : VGPR0 = {lanes0-15: A[m][k+0], lanes16-31: A[m][k+2]}
//            VGPR1 = {lanes0-15: A[m][k+1], lanes16-31: A[m][k+3]}
//   B 4x16 : VGPR0 = {lanes0-15: B[k+0][n], lanes16-31: B[k+2][n]}
//            VGPR1 = {lanes0-15: B[k+1][n], lanes16-31: B[k+3][n]}
//   C 16x16: VGPR r = {lanes0-15: C[r][n], lanes16-31: C[r+8][n]}
// ---------------------------------------------------------------------------
__global__ void mlp_gemm_wmma(const float* __restrict__ A,
                              const float* __restrict__ W,
                              const float* __restrict__ bias,
                              float* __restrict__ out,
                              int do_relu) {
    const int lane = threadIdx.x;        // 0..31, full wave active (EXEC all 1s)
    const int n0   = blockIdx.x * 16;    // column tile
    const int m    = lane & 15;          // A row for this lane
    const int half = lane >> 4;          // 0: K 0..1, 1: K 2..3
    const int n    = lane & 15;          // B/C column for this lane

    v8f c = {};
    for (int k = 0; k < D_; k += 4) {
        const int ka = k + half * 2;
        v2f a, b;
        a.x = A[m * D_ + ka];
        a.y = A[m * D_ + ka + 1];
        b.x = W[(size_t)ka * D_ + n0 + n];
        b.y = W[(size_t)(ka + 1) * D_ + n0 + n];
        // 8 args: (neg_a, A, neg_b, B, c_mod, C, reuse_a, reuse_b)
        c = __builtin_amdgcn_wmma_f32_16x16x4_f32(
            /*neg_a=*/false, a, /*neg_b=*/false, b,
            /*c_mod=*/(short)0, c, /*reuse_a=*/false, /*reuse_b=*/false);
    }

    const float bv = bias[n0 + n];
#pragma unroll
    for (int r = 0; r < 8; ++r) {
        float v = c[r] + bv;
        if (do_relu) v = fmaxf(v, 0.0f);
        const int M = r + half * 8;
        out[M * D_ + n0 + n] = v;
    }
}

// ---------------------------------------------------------------------------
// Per-row running count of pseudo tokens: pidx[b,s] = clip(cumsum(mask)-1,0,P-1)
// 32 rows -> 32 threads, trivial cost (tokens fit in L2).
// ---------------------------------------------------------------------------
__global__ void pidx_scan(const int* __restrict__ inputs, int* __restrict__ pidx) {
    const int b = threadIdx.x;
    if (b >= B_) return;
    int cnt = 0;
    for (int s = 0; s < S_; ++s) {
        const int tok = inputs[b * S_ + s];
        int idx = 0;
        if (tok == PID_) {
            idx = cnt < (P_ - 1) ? cnt : (P_ - 1);
            ++cnt;
        }
        pidx[b * S_ + s] = idx;
    }
}

// ---------------------------------------------------------------------------
// Fused: embedding gather / prompt scatter-select + pos add + LayerNorm.
// One block per (b,s) token; 256 threads * float4 = D=1024 elements.
// Single HBM pass over the 128MB in / 128MB out => ~11us roofline @ 23.3TB/s.
// ---------------------------------------------------------------------------
__global__ void embed_ln_fused(const int* __restrict__ inputs,
                               const float* __restrict__ wte,
                               const float* __restrict__ wpe,
                               const float* __restrict__ rep,
                               const int* __restrict__ pidx,
                               const float* __restrict__ gamma,
                               const float* __restrict__ beta,
                               float* __restrict__ out) {
    const int bs = blockIdx.x;           // 0 .. B*S-1
    const int s  = bs & (S_ - 1);
    const int t  = threadIdx.x;          // 0..255
    const int d  = t * 4;

    const int tok = inputs[bs];
    const float* src = (tok == PID_)
        ? (rep + (size_t)pidx[bs] * D_ + d)
        : (wte + (size_t)tok * D_ + d);

    float4 h = *(const float4*)src;
    const float4 p = *(const float4*)(wpe + (size_t)s * D_ + d);
    h.x += p.x; h.y += p.y; h.z += p.z; h.w += p.w;

    // block reduction for mean / mean-of-squares (8 wave32s)
    float sum = h.x + h.y + h.z + h.w;
    float sq  = h.x * h.x + h.y * h.y + h.z * h.z + h.w * h.w;
#pragma unroll
    for (int off = 16; off > 0; off >>= 1) {
        sum += __shfl_down(sum, off, 32);
        sq  += __shfl_down(sq,  off, 32);
    }
    __shared__ float s_sum[8], s_sq[8];
    const int wave = t >> 5, ln = t & 31;
    if (ln == 0) { s_sum[wave] = sum; s_sq[wave] = sq; }
    __syncthreads();
    if (wave == 0) {
        float a = (ln < 8) ? s_sum[ln] : 0.0f;
        float q = (ln < 8) ? s_sq[ln]  : 0.0f;
#pragma unroll
        for (int off = 4; off > 0; off >>= 1) {
            a += __shfl_down(a, off, 32);
            q += __shfl_down(q, off, 32);
        }
        if (ln == 0) { s_sum[0] = a; s_sq[0] = q; }
    }
    __syncthreads();

    const float mean = s_sum[0] * (1.0f / D_);
    const float var  = s_sq[0] * (1.0f / D_) - mean * mean;
    const float inv  = rsqrtf(var + EPS_);

    const float4 g = *(const float4*)(gamma + d);
    const float4 bt = *(const float4*)(beta + d);
    float4 o;
    o.x = (h.x - mean) * inv * g.x + bt.x;
    o.y = (h.y - mean) * inv * g.y + bt.y;
    o.z = (h.z - mean) * inv * g.z + bt.z;
    o.w = (h.w - mean) * inv * g.w + bt.w;
    *(float4*)(out + (size_t)bs * D_ + d) = o;
}

extern "C" void kernel_launch(void* const* d_in, const int* in_sizes, int n_in,
                              void* d_out, int out_size, void* d_ws, size_t ws_size,
                              hipStream_t stream) {
    const int*   inputs     = (const int*)  d_in[0];   // [B,S]
    const float* wte        = (const float*)d_in[1];   // [V,D]
    const float* wpe        = (const float*)d_in[2];   // [S,D]
    const float* prompt_emb = (const float*)d_in[3];   // [P,D]
    const float* W1         = (const float*)d_in[4];   // [D,D]
    const float* b1         = (const float*)d_in[5];   // [D]
    const float* W2         = (const float*)d_in[6];   // [D,D]
    const float* b2         = (const float*)d_in[7];   // [D]
    const float* gamma      = (const float*)d_in[8];   // [D]
    const float* beta       = (const float*)d_in[9];   // [D]
    float* out = (float*)d_out;

    // workspace layout
    char* ws = (char*)d_ws;
    float* hidden = (float*)ws;                                   // 16*1024 f32 = 64KB
    float* rep    = (float*)(ws + (size_t)P_ * D_ * sizeof(float));       // 64KB
    int*   pidx   = (int*)  (ws + (size_t)2 * P_ * D_ * sizeof(float));   // 128KB

    // prompt MLP: hidden = relu(prompt_emb @ W1 + b1); rep = hidden @ W2 + b2
    mlp_gemm_wmma<<<D_ / 16, 32, 0, stream>>>(prompt_emb, W1, b1, hidden, 1);
    mlp_gemm_wmma<<<D_ / 16, 32, 0, stream>>>(hidden,     W2, b2, rep,    0);

    // per-row pseudo-token running index
    pidx_scan<<<1, 32, 0, stream>>>(inputs, pidx);

    // fused gather/scatter + pos-embed + LayerNorm
    embed_ln_fused<<<B_ * S_, 256, 0, stream>>>(inputs, wte, wpe, rep, pidx,
                                                gamma, beta, out);
}